// HausdorffDist_27015344292557
// MI455X (gfx1250) — compile-verified
//
#include <hip/hip_runtime.h>
#include <hip/hip_bf16.h>
#include <float.h>

typedef __attribute__((ext_vector_type(2))) float v2f;
typedef __attribute__((ext_vector_type(8))) float v8f;

// ---------------------------------------------------------------------------
// Zero the per-batch running-max accumulators (workspace is poisoned 0xAA).
// Distances are >= 0, so 0.0f is a safe identity for the max reduction.
// ---------------------------------------------------------------------------
__global__ void haus_init(float* __restrict__ bmax, int Bn) {
    int i = threadIdx.x;
    if (i < Bn) bmax[i] = 0.0f;
}

// ---------------------------------------------------------------------------
// Main kernel: one wave per (batch, 32-column adv strip) = two 16-col tiles.
// For each 16-row ori tile, ONE A fragment feeds TWO V_WMMA_F32_16X16X4_F32:
//   D[n,m] = xx_n - 2 * <ori_n, adv_m>   (K=4: [ox,oy,oz,xx] x [-2ax,-2ay,-2az,1])
// This amortizes the A-fragment load + xx/cndmask prep (the VALU critical
// path) over two matrix ops. Running min over rows is kept in the 8-VGPR f32
// accumulator layout, combined across lane halves, biased by yy_m, max-reduced
// over columns, and atomically max'ed into bmax[b] (int bit-pattern compare,
// valid for non-negative floats).
// ---------------------------------------------------------------------------
__global__ __launch_bounds__(256) void haus_main(const float* __restrict__ adv,
                                                 const float* __restrict__ ori,
                                                 float* __restrict__ bmax,
                                                 int Bn, int K) {
    const int lane = threadIdx.x & 31;
    const int half = lane >> 4;     // 0: lanes 0-15 (K=0,1), 1: lanes 16-31 (K=2,3)
    const int l16  = lane & 15;

    const int wavesPerBlock = blockDim.x >> 5;
    const int gw   = blockIdx.x * wavesPerBlock + (threadIdx.x >> 5);
    const int stripsPerBatch = K >> 5;                    // 32 columns per wave
    const int b     = gw / stripsPerBatch;
    const int strip = gw - b * stripsPerBatch;
    if (b >= Bn) return;                                  // uniform per wave

    // ---- Two B fragments: columns m0 = strip*32 + l16, m1 = m0 + 16 ----
    const int col0 = (strip << 5) + l16;
    const float* ap0 = adv + ((size_t)b * K + col0) * 3;
    const float* ap1 = ap0 + 16 * 3;

    const float ax0 = ap0[0], ay0 = ap0[1], az0 = ap0[2];
    const float ax1 = ap1[0], ay1 = ap1[1], az1 = ap1[2];
    const float yy0 = ax0 * ax0 + ay0 * ay0 + az0 * az0;
    const float yy1 = ax1 * ax1 + ay1 * ay1 + az1 * az1;

    v2f bfrag0, bfrag1;
    bfrag0.x = half ? (-2.0f * az0) : (-2.0f * ax0);   // K=2 : K=0
    bfrag0.y = half ? 1.0f          : (-2.0f * ay0);   // K=3 : K=1
    bfrag1.x = half ? (-2.0f * az1) : (-2.0f * ax1);
    bfrag1.y = half ? 1.0f          : (-2.0f * ay1);

    v8f rmin0, rmin1;
#pragma unroll
    for (int i = 0; i < 8; ++i) { rmin0[i] = FLT_MAX; rmin1[i] = FLT_MAX; }

    const float* obase = ori + (size_t)b * K * 3;
    for (int r = 0; r < K; r += 16) {
        // A fragment: row n = ori point, [ox, oy, oz, xx] -- built ONCE,
        // consumed by BOTH column tiles.
        const float* op = obase + (size_t)(r + l16) * 3;
        const float ox = op[0], oy = op[1], oz = op[2];
        const float xx = ox * ox + oy * oy + oz * oz;

        v2f afrag;
        afrag.x = half ? oz : ox;                   // K=2 : K=0
        afrag.y = half ? xx : oy;                   // K=3 : K=1

        v8f c0 = {}, c1 = {};
        c0 = __builtin_amdgcn_wmma_f32_16x16x4_f32(
                false, afrag, false, bfrag0, (short)0, c0, false, false);
        c1 = __builtin_amdgcn_wmma_f32_16x16x4_f32(
                false, afrag, false, bfrag1, (short)0, c1, false, false);

#pragma unroll
        for (int i = 0; i < 8; ++i) {
            rmin0[i] = fminf(rmin0[i], c0[i]);
            rmin1[i] = fminf(rmin1[i], c1[i]);
        }
    }

    // min over the 8 rows each lane holds, per column tile
    float m0 = fminf(fminf(fminf(rmin0[0], rmin0[1]), fminf(rmin0[2], rmin0[3])),
                     fminf(fminf(rmin0[4], rmin0[5]), fminf(rmin0[6], rmin0[7])));
    float m1 = fminf(fminf(fminf(rmin1[0], rmin1[1]), fminf(rmin1[2], rmin1[3])),
                     fminf(fminf(rmin1[4], rmin1[5]), fminf(rmin1[6], rmin1[7])));

    // combine lane halves: rows 0-7 (lanes 0-15) with rows 8-15 (lanes 16-31)
    m0 = fminf(m0, __shfl_xor(m0, 16, 32));
    m1 = fminf(m1, __shfl_xor(m1, 16, 32));

    // add per-column bias -> true min_n ||ori_n - adv_m||^2, then local max
    float m = fmaxf(m0 + yy0, m1 + yy1);

    // max over the 16 column slots of this wave
#pragma unroll
    for (int off = 8; off >= 1; off >>= 1)
        m = fmaxf(m, __shfl_xor(m, off, 32));

    if (lane == 0)
        atomicMax((int*)(bmax + b), __float_as_int(m));   // m >= 0 -> monotone
}

// ---------------------------------------------------------------------------
// Final: mean over batch of bmax[b] * weights[b]  -> d_out[0]
// ---------------------------------------------------------------------------
__global__ void haus_final(const float* __restrict__ bmax,
                           const float* __restrict__ w,
                           float* __restrict__ out, int Bn) {
    int i = threadIdx.x;
    float acc = (i < Bn) ? bmax[i] * w[i] : 0.0f;
#pragma unroll
    for (int off = 16; off >= 1; off >>= 1)
        acc += __shfl_xor(acc, off, 32);
    if (i == 0) out[0] = acc / (float)Bn;
}

extern "C" void kernel_launch(void* const* d_in, const int* in_sizes, int n_in,
                              void* d_out, int out_size, void* d_ws, size_t ws_size,
                              hipStream_t stream) {
    const float* adv = (const float*)d_in[0];   // [B, K, 3]
    const float* ori = (const float*)d_in[1];   // [B, K, 3]
    const float* w   = (const float*)d_in[2];   // [B]

    const int Bn = in_sizes[2];
    const int K  = in_sizes[0] / (3 * Bn);

    float* bmax = (float*)d_ws;                 // Bn floats of scratch

    haus_init<<<1, 32, 0, stream>>>(bmax, Bn);

    const int stripsPerBatch = K >> 5;          // 32 columns per wave
    const int totalWaves     = Bn * stripsPerBatch;
    const int wavesPerBlock  = 8;               // 256 threads = 8 wave32
    const int blocks         = (totalWaves + wavesPerBlock - 1) / wavesPerBlock;

    haus_main<<<blocks, wavesPerBlock * 32, 0, stream>>>(adv, ori, bmax, Bn, K);

    haus_final<<<1, 32, 0, stream>>>(bmax, w, (float*)d_out, Bn);
}